// Ergodicity_Loss_12395275616936
// MI455X (gfx1250) — compile-verified
//
#include <hip/hip_runtime.h>
#include <hip/hip_bf16.h>

// ---------------------------------------------------------------------------
// Ergodicity loss on gfx1250:
//   M_b[i][j] = sum_p cos(i*pi*x0_p) * cos(j*pi*x1_p)   (rank-1 accumulation)
// mapped to V_WMMA_F32_16X16X4_F32 (full-f32 matrix core path), followed by a
// tiny deterministic finalize pass computing the weighted MSE loss.
//
// Round-1 fixes vs round-0:
//  * no lane predication (padded modes 10..15 carry garbage that is never
//    read by the finalize pass) -> no exec-mask save/restore, no branches
//  * inner loop of 8 WMMA steps per timestep uses immediate-offset b128
//    loads from a single per-lane base pointer -> loads batch & overlap
// ---------------------------------------------------------------------------

typedef float v2f __attribute__((ext_vector_type(2)));
typedef float v4f __attribute__((ext_vector_type(4)));
typedef float v8f __attribute__((ext_vector_type(8)));

#define ERG_T        512
#define ERG_B        16
#define ERG_N        32
#define ERG_PPB      (ERG_T * ERG_N)       // 16384 points per batch
#define ERG_WGPB     8                     // blocks (slices) per batch
#define ERG_PPBLK    (ERG_PPB / ERG_WGPB)  // 2048 points per block
#define ERG_PPW      (ERG_PPBLK / 4)       // 512 points per wave
#define ERG_TSTEPS   (ERG_PPW / ERG_N)     // 16 timesteps per wave

// cos(pi*m*x) == v_cos_f32(0.5*m*x)   (v_cos input is in revolutions)
__device__ __forceinline__ float pi_cos(float half_m, float x) {
    return __builtin_amdgcn_cosf(half_m * x);
}

__global__ __launch_bounds__(128)
void erg_basis_gemm(const float* __restrict__ x, float* __restrict__ part) {
    const int g    = blockIdx.x;       // 0..127
    const int b    = g >> 3;           // batch 0..15
    const int s    = g & 7;            // slice 0..7
    const int tid  = threadIdx.x;
    const int wave = tid >> 5;         // 0..3 (wave32)
    const int lane = tid & 31;
    const int mode = lane & 15;        // mode index for both A row and B col
    const int hi   = lane >> 4;        // selects K-pair {0,1} vs {2,3}

    const float half_m = 0.5f * (float)mode;   // modes 10..15 = harmless padding

    // first timestep handled by this wave
    const int t0 = (s * ERG_PPBLK + wave * ERG_PPW) >> 5;

    // per-lane base: x[t0][b][n = 2*hi][0]; advances 16*32*2 floats per timestep
    const float* px = x + (((size_t)t0 * ERG_B + b) * ERG_N) * 2 + (hi << 2);

    v8f acc = {};

    for (int tt = 0; tt < ERG_TSTEPS; ++tt) {
        #pragma unroll
        for (int c = 0; c < 8; ++c) {
            // points n = 4c + 2*hi, 4c + 2*hi + 1 of this timestep
            const v4f xv = *(const v4f*)(px + (c << 3));  // {x0,x1, x0',x1'}

            v2f A, Bm;
            A.x  = pi_cos(half_m, xv.x);   // A[mode][kk0]   = cos(pi*m*x0)
            Bm.x = pi_cos(half_m, xv.y);   // B[kk0][mode]   = cos(pi*m*x1)
            A.y  = pi_cos(half_m, xv.z);   // A[mode][kk0+1]
            Bm.y = pi_cos(half_m, xv.w);   // B[kk0+1][mode]

            // D = A(16x4) * B(4x16) + C, all f32
            acc = __builtin_amdgcn_wmma_f32_16x16x4_f32(
                /*neg_a=*/false, A, /*neg_b=*/false, Bm,
                /*c_mod=*/(short)0, acc, /*reuse_a=*/false, /*reuse_b=*/false);
        }
        px += ERG_B * ERG_N * 2;   // next timestep (+4096 bytes)
    }

    // Deterministic cross-wave reduction via LDS (no atomics).
    // D layout: VGPR r holds D[M = r + 8*hi][N = mode]
    __shared__ float red[4 * 256];
    float* my = &red[wave * 256];
    #pragma unroll
    for (int r = 0; r < 8; ++r) {
        const int M = r + (hi << 3);
        my[M * 16 + mode] = acc[r];
    }
    __syncthreads();
    for (int idx = tid; idx < 256; idx += 128) {
        const float s4 = red[idx] + red[256 + idx] + red[512 + idx] + red[768 + idx];
        part[g * 256 + idx] = s4;  // per-block partial tile
    }
}

__global__ __launch_bounds__(256)
void erg_finalize(const float* __restrict__ part, float* __restrict__ out) {
    const float PI = 3.14159265358979323846f;
    const int tid = threadIdx.x;
    float sum = 0.f;

    for (int idx = tid; idx < ERG_B * 100; idx += 256) {
        const int b = idx / 100;
        const int k = idx % 100;
        const int i = k / 10;    // K[:,0]
        const int j = k % 10;    // K[:,1]  (last dim -> drives nf)

        // sum the 8 slice partials in fixed order (deterministic)
        float M = 0.f;
        #pragma unroll
        for (int s = 0; s < 8; ++s)
            M += part[(b * 8 + s) * 256 + i * 16 + j];

        const float nf = (j != 0) ? 0.70710678118654752440f : 1.0f;  // sqrt(1/2)
        const float coeff = M * (1.0f / (float)ERG_PPB) / nf;

        // characteristic function of uniform density:
        // term(m!=0) purely imaginary: Im = (1-(-1)^m)/(m*pi); term(0)=1
        const float Rei = (i == 0) ? 1.f : 0.f;
        const float Rej = (j == 0) ? 1.f : 0.f;
        const float Imi = (i == 0) ? 0.f : ((i & 1) ? 2.f / ((float)i * PI) : 0.f);
        const float Imj = (j == 0) ? 0.f : ((j & 1) ? 2.f / ((float)j * PI) : 0.f);
        const float cd  = (Rei * Rej - Imi * Imj) / nf;

        const float base = 1.f + PI * PI * (float)(i * i + j * j);
        const float nw   = 100.f / (base * __builtin_sqrtf(base));   // base^-1.5 * 100

        const float d = nw * (coeff - cd);
        sum += d * d;
    }

    __shared__ float sred[256];
    sred[tid] = sum;
    __syncthreads();
    for (int off = 128; off > 0; off >>= 1) {
        if (tid < off) sred[tid] += sred[tid + off];
        __syncthreads();
    }
    if (tid == 0)
        out[0] = sred[0] * (1.0f / (float)(ERG_B * 100));  // mean over [16,100]
}

extern "C" void kernel_launch(void* const* d_in, const int* in_sizes, int n_in,
                              void* d_out, int out_size, void* d_ws, size_t ws_size,
                              hipStream_t stream) {
    const float* x = (const float*)d_in[0];    // [512,16,32,2] f32
    float* out     = (float*)d_out;            // scalar f32
    float* part    = (float*)d_ws;             // 128 blocks * 256 f32 = 128 KB

    erg_basis_gemm<<<dim3(ERG_B * ERG_WGPB), dim3(128), 0, stream>>>(x, part);
    erg_finalize<<<dim3(1), dim3(256), 0, stream>>>(part, out);
}